// AdaptiveCrossHadamard_22376779612367
// MI455X (gfx1250) — compile-verified
//
#include <hip/hip_runtime.h>

typedef float v2f __attribute__((ext_vector_type(2)));
typedef float v8f __attribute__((ext_vector_type(8)));
typedef int   v4i __attribute__((ext_vector_type(4)));

#define C1     256
#define BATCH  16
#define HW     4096
#define COUT   752
#define CS     32
#define CSE    496
#define EPSV   1e-5f
#define XROW   288   // interleaved X row stride (dwords); 288 % 64 == 32 -> halves hit disjoint banks
#define WROW   36    // W row stride (dwords); conflict-free, 16B aligned

// gfx1250 async global->LDS staging. Signatures probe-confirmed:
//   b128: (v4i* src, __shared__ v4i* dst, int offset, int cpol)
//   b32 : (int* src, __shared__ int* dst, int offset, int cpol)
#if defined(__has_builtin)
#if __has_builtin(__builtin_amdgcn_global_load_async_to_lds_b128) && \
    __has_builtin(__builtin_amdgcn_global_load_async_to_lds_b32)  && \
    __has_builtin(__builtin_amdgcn_s_wait_asynccnt)
#define USE_ASYNC_LDS 1
#endif
#endif
#ifdef USE_ASYNC_LDS
typedef __attribute__((address_space(3))) v4i as3_v4i;
typedef __attribute__((address_space(3))) int as3_int;
#endif

// ---------------------------------------------------------------------------
// Kernel 0: zero the pooled-sum accumulator (16*256 floats)
// ---------------------------------------------------------------------------
__global__ void zero_pool_kernel(float* __restrict__ pool) {
    pool[blockIdx.x * 256 + threadIdx.x] = 0.0f;
}

// ---------------------------------------------------------------------------
// Kernel 1: per-pixel 256x256 matmul via V_WMMA_F32_16X16X4_F32, fused
// bias + BatchNorm(eval), plus per-(b,channel) pooled-sum accumulation.
// Block: 512 threads = 16 waves. Each wave owns 16 output channels (M=16),
// the block covers M=256 x N=128 pixels for one batch image.
// ---------------------------------------------------------------------------
__global__ __launch_bounds__(512) void fc_bn_pool_kernel(
    const float* __restrict__ x,    // [B, 256, 4096]
    const float* __restrict__ w,    // [256, 256]
    const float* __restrict__ bfc,  // [256]
    const float* __restrict__ gam, const float* __restrict__ bet,
    const float* __restrict__ mu,  const float* __restrict__ var,
    float* __restrict__ out,        // [B, 752, 4096] (first 256 ch written)
    float* __restrict__ pool)       // [B, 256] sums
{
    __shared__ float Wl[256 * WROW];   // W chunk [256][32], row stride 36
    __shared__ float Xl[16 * XROW];    // X chunk, K-pair interleaved:
                                       //   Xl[(k>>1)*XROW + 2*n + (k&1)]
    __shared__ float scl[256];
    __shared__ float off[256];

    const int tid   = threadIdx.x;
    const int b     = blockIdx.y;
    const int hw0   = blockIdx.x * 128;
    const int wave  = tid >> 5;
    const int lane  = tid & 31;
    const int lr    = lane & 15;     // lane within half
    const int hh    = lane >> 4;     // which 16-lane half
    const int obase = wave * 16;     // this wave's output-channel base

    // Fused per-channel scale/offset: y = acc*scl + off
    if (tid < 256) {
        float s = gam[tid] * rsqrtf(var[tid] + EPSV);
        scl[tid] = s;
        off[tid] = (bfc[tid] - mu[tid]) * s + bet[tid];
    }

    v8f acc[8];
    const v8f zero = {0.f, 0.f, 0.f, 0.f, 0.f, 0.f, 0.f, 0.f};
#pragma unroll
    for (int j = 0; j < 8; ++j) acc[j] = zero;

    const float* xb = x + (size_t)b * C1 * HW;

    for (int k0 = 0; k0 < 256; k0 += 32) {
        __syncthreads();
        // ---- stage W chunk: rows o=0..255, cols k0..k0+31 ----
        {
            const int r0 = tid >> 3;
            const int q  = (tid & 7) << 2;
#pragma unroll
            for (int it = 0; it < 4; ++it) {
                const int row = r0 + (it << 6);
#ifdef USE_ASYNC_LDS
                __builtin_amdgcn_global_load_async_to_lds_b128(
                    (v4i*)(w + row * 256 + k0 + q),
                    (as3_v4i*)&Wl[row * WROW + q], 0, 0);
#else
                const float4 wv = *(const float4*)(w + row * 256 + k0 + q);
                *(float4*)&Wl[row * WROW + q] = wv;
#endif
            }
        }
        // ---- stage X chunk (K-pair interleaved): rows k=0..31 ----
        {
            const int r0 = tid >> 5;
            const int q  = (tid & 31) << 2;
#pragma unroll
            for (int it = 0; it < 2; ++it) {
                const int k = r0 + (it << 4);
                const float* gsrc = xb + (size_t)(k0 + k) * HW + hw0 + q;
                float* dst = &Xl[(k >> 1) * XROW + (k & 1)];
#ifdef USE_ASYNC_LDS
#pragma unroll
                for (int e = 0; e < 4; ++e)
                    __builtin_amdgcn_global_load_async_to_lds_b32(
                        (int*)(gsrc + e),
                        (as3_int*)&dst[(q + e) << 1], 0, 0);
#else
                const float4 xv = *(const float4*)gsrc;
                dst[(q + 0) << 1] = xv.x;
                dst[(q + 1) << 1] = xv.y;
                dst[(q + 2) << 1] = xv.z;
                dst[(q + 3) << 1] = xv.w;
#endif
            }
        }
        // prefetch next X chunk rows into cache (speculative)
        if (k0 + 32 < 256)
            __builtin_prefetch(xb + (size_t)(k0 + 32 + (tid >> 5)) * HW + hw0, 0, 0);
#ifdef USE_ASYNC_LDS
        __builtin_amdgcn_s_wait_asynccnt(0);
#endif
        __syncthreads();

        // ---- 8 K-steps of K=4 per chunk; 8 WMMA (N-subtiles) per step ----
#pragma unroll
        for (int kk = 0; kk < 32; kk += 4) {
            // A fragment: lanes 0-15 take K=kk,kk+1; lanes 16-31 take K=kk+2,kk+3
            const v2f a = *(const v2f*)&Wl[(obase + lr) * WROW + kk + (hh << 1)];
            // B fragments: interleaved pair {X[krow][n], X[krow+1][n]} = one b64
            const float* xrow = &Xl[((kk >> 1) + hh) * XROW + (lr << 1)];
#pragma unroll
            for (int j = 0; j < 8; ++j) {
                const v2f bb = *(const v2f*)&xrow[j << 5];
                acc[j] = __builtin_amdgcn_wmma_f32_16x16x4_f32(
                    false, a, false, bb, (short)0, acc[j], false, false);
            }
        }
    }

    // ---- epilogue: BN, store, pooled partial sums ----
    const size_t outB = (size_t)b * COUT * HW;
#pragma unroll
    for (int r = 0; r < 8; ++r) {
        const int oo = obase + r + (hh << 3);    // C-layout: lanes16-31 -> M+8
        const float s = scl[oo];
        const float o = off[oo];
        float ps = 0.f;
#pragma unroll
        for (int j = 0; j < 8; ++j) {
            const float v = acc[j][r] * s + o;
            out[outB + (size_t)oo * HW + hw0 + (j << 4) + lr] = v;
            ps += v;
        }
        // reduce the 16 lanes of this half (stays within half for masks<16)
#pragma unroll
        for (int m = 1; m < 16; m <<= 1) ps += __shfl_xor(ps, m, 32);
        if (lr == 0) atomicAdd(&pool[b * 256 + oo], ps);
    }
}

// ---------------------------------------------------------------------------
// Kernel 2: ECA conv1d(k=5,pad=2) over channel axis of pooled means, then
// stable top-32 selection (rank counting reproduces jax.lax.top_k order).
// One block per batch image.
// ---------------------------------------------------------------------------
__global__ __launch_bounds__(256) void eca_topk_kernel(
    const float* __restrict__ pool, const float* __restrict__ ew,
    int* __restrict__ topk)
{
    __shared__ float pm[256];
    __shared__ float lg[256];
    const int b = blockIdx.x;
    const int c = threadIdx.x;

    pm[c] = pool[b * 256 + c] * (1.0f / (float)HW);
    __syncthreads();

    float a = 0.f;
#pragma unroll
    for (int d = 0; d < 5; ++d) {
        const int cc = c - 2 + d;
        if (cc >= 0 && cc < 256) a += ew[d] * pm[cc];
    }
    lg[c] = a;
    __syncthreads();

    const float mine = lg[c];
    int rank = 0;
    for (int d = 0; d < 256; ++d) {
        const float o = lg[d];
        rank += (o > mine) || (o == mine && d < c);
    }
    if (rank < CS) topk[b * CS + rank] = c;
}

// ---------------------------------------------------------------------------
// Kernel 3: gather 32 selected channels to LDS, emit all 496 (i<j) Hadamard
// products with fused bn_s. Block covers 256 pixels of one batch image.
// Output channels are write-only -> non-temporal stores (keep y in L2).
// ---------------------------------------------------------------------------
__global__ __launch_bounds__(256) void hadamard_kernel(
    const float* __restrict__ yRO,   // alias of out (first 256 channels)
    const int* __restrict__ topk,
    const float* __restrict__ gam, const float* __restrict__ bet,
    const float* __restrict__ mu,  const float* __restrict__ var,
    float* __restrict__ out)
{
    __shared__ float sel[CS][256];
    __shared__ float ss[CSE];
    __shared__ float so[CSE];
    __shared__ unsigned char hia[CSE];
    __shared__ unsigned char hja[CSE];
    __shared__ int idxs[CS];

    const int tid = threadIdx.x;
    const int b   = blockIdx.y;
    const int hw0 = blockIdx.x * 256;

    if (tid < CS) idxs[tid] = topk[b * CS + tid];

    // triu_indices(32, k=1) pair table + fused bn_s params
    for (int p = tid; p < CSE; p += 256) {
        int offp = 0, i = 0;
        for (; i < 31; ++i) {
            const int cnt = 31 - i;
            if (p < offp + cnt) break;
            offp += cnt;
        }
        hia[p] = (unsigned char)i;
        hja[p] = (unsigned char)(i + 1 + (p - offp));
        const float s = gam[p] * rsqrtf(var[p] + EPSV);
        ss[p] = s;
        so[p] = bet[p] - mu[p] * s;
    }
    __syncthreads();

    const size_t base = (size_t)b * COUT * HW + hw0;
#pragma unroll 4
    for (int i = 0; i < CS; ++i)
        sel[i][tid] = yRO[base + (size_t)idxs[i] * HW + tid];
    __syncthreads();

    for (int p = 0; p < CSE; ++p) {
        const float v = sel[hia[p]][tid] * sel[hja[p]][tid];
        __builtin_nontemporal_store(v * ss[p] + so[p],
                                    &out[base + (size_t)(256 + p) * HW + tid]);
    }
}

// ---------------------------------------------------------------------------
extern "C" void kernel_launch(void* const* d_in, const int* in_sizes, int n_in,
                              void* d_out, int out_size, void* d_ws, size_t ws_size,
                              hipStream_t stream) {
    (void)in_sizes; (void)n_in; (void)out_size; (void)ws_size;
    const float* x   = (const float*)d_in[0];
    const float* wfc = (const float*)d_in[1];
    const float* bfc = (const float*)d_in[2];
    const float* bxg = (const float*)d_in[3];
    const float* bxb = (const float*)d_in[4];
    const float* bxm = (const float*)d_in[5];
    const float* bxv = (const float*)d_in[6];
    const float* ew  = (const float*)d_in[7];
    const float* bsg = (const float*)d_in[8];
    const float* bsb = (const float*)d_in[9];
    const float* bsm = (const float*)d_in[10];
    const float* bsv = (const float*)d_in[11];

    float* out  = (float*)d_out;
    float* pool = (float*)d_ws;                       // 16*256 floats
    int*   topk = (int*)((char*)d_ws + 16 * 256 * 4); // 16*32 ints

    zero_pool_kernel<<<dim3(BATCH), dim3(256), 0, stream>>>(pool);
    fc_bn_pool_kernel<<<dim3(HW / 128, BATCH), dim3(512), 0, stream>>>(
        x, wfc, bfc, bxg, bxb, bxm, bxv, out, pool);
    eca_topk_kernel<<<dim3(BATCH), dim3(256), 0, stream>>>(pool, ew, topk);
    hadamard_kernel<<<dim3(HW / 256, BATCH), dim3(256), 0, stream>>>(
        out, topk, bsg, bsb, bsm, bsv, out);
}